// H100GCN_11665131176367
// MI455X (gfx1250) — compile-verified
//
#include <hip/hip_runtime.h>
#include <math.h>

#define N_NODES 100000
#define N_EDGES 1600000
#define IN_DIM  128
#define HID     64

typedef float v2f __attribute__((ext_vector_type(2)));
typedef float v8f __attribute__((ext_vector_type(8)));

// ---------------- degree / dinv ----------------
__global__ void gcn_deg_kernel(const int* __restrict__ dst, float* __restrict__ deg) {
    int e = blockIdx.x * blockDim.x + threadIdx.x;
    if (e < N_EDGES) atomicAdd(&deg[dst[e]], 1.0f);
}

__global__ void gcn_dinv_kernel(float* __restrict__ deg) {
    int i = blockIdx.x * blockDim.x + threadIdx.x;
    if (i < N_NODES) deg[i] = rsqrtf(deg[i] + 1.0f);
}

// ---------------- WMMA f32 GEMM: Hout[N,64] = X[N,K] @ W[K,64] ----------------
// blockDim = 256 (8 waves). Each wave computes a 16-row x 64-col stripe using
// four 16x16 f32 accumulators and V_WMMA_F32_16X16X4_F32 along K.
// A rows are CLAMPED (not predicated): tail-row garbage only reaches D rows
// that the guarded store skips, and the hot loop keeps EXEC all-1s.
template <int K>
__global__ __launch_bounds__(256)
void gcn_gemm_wmma_kernel(const float* __restrict__ X, const float* __restrict__ W,
                          float* __restrict__ Hout) {
    const int wave = threadIdx.x >> 5;
    const int lane = threadIdx.x & 31;
    const int half = lane >> 4;       // 0: K pair {k,k+1}, 1: {k+2,k+3}
    const int l16  = lane & 15;
    const int row0 = (blockIdx.x * 8 + wave) * 16;

    v8f acc[4];
#pragma unroll
    for (int t = 0; t < 4; ++t)
#pragma unroll
        for (int j = 0; j < 8; ++j) acc[t][j] = 0.0f;

    int arow = row0 + l16;
    if (arow > N_NODES - 1) arow = N_NODES - 1;   // clamp: no divergent guard in loop
    const float* __restrict__ Arow = X + (size_t)arow * K + 2 * half;

    for (int k = 0; k < K; k += 4) {
        // A 16x4: lanes 0-15 hold K=k..k+1, lanes 16-31 hold K=k+2..k+3 (ISA 7.12.2)
        v2f a;
        a.x = Arow[k];
        a.y = Arow[k + 1];
#pragma unroll
        for (int t = 0; t < 4; ++t) {
            // B 4x16 (cols t*16..t*16+15): same K striping over lane halves
            const float* q = W + (size_t)(k + 2 * half) * HID + t * 16 + l16;
            v2f b; b.x = q[0]; b.y = q[HID];
            acc[t] = __builtin_amdgcn_wmma_f32_16x16x4_f32(
                false, a, false, b, (short)0, acc[t], false, false);
        }
    }

    // C/D layout: VGPR j, lanes 0-15 -> row row0+j, lanes 16-31 -> row row0+8+j
#pragma unroll
    for (int t = 0; t < 4; ++t)
#pragma unroll
        for (int j = 0; j < 8; ++j) {
            int r = row0 + j + half * 8;
            if (r < N_NODES) Hout[(size_t)r * HID + t * 16 + l16] = acc[t][j];
        }
}

// ---------------- edge scatter: agg[dst] += h[src] * dinv[src]*dinv[dst] ----------------
// One thread per (edge, 4 features): float4 gather of h (global_load_b128) +
// 4 f32 atomics. 16 consecutive threads share one edge -> index/dinv loads
// broadcast from cache; h/agg (25.6MB each) live in the 192MB L2.
__global__ __launch_bounds__(256)
void gcn_scatter_kernel(const int* __restrict__ src, const int* __restrict__ dst,
                        const float* __restrict__ dinv, const float* __restrict__ h,
                        float* __restrict__ agg) {
    long long t = (long long)blockIdx.x * blockDim.x + threadIdx.x;
    if (t >= (long long)N_EDGES * (HID / 4)) return;
    int e  = (int)(t >> 4);
    int f4 = ((int)t & 15) * 4;
    int s = src[e];
    int d = dst[e];
    float coef = dinv[s] * dinv[d];
    const float4 hv = *(const float4*)(h + (size_t)s * HID + f4);
    float* ap = agg + (size_t)d * HID + f4;
    atomicAdd(ap + 0, hv.x * coef);
    atomicAdd(ap + 1, hv.y * coef);
    atomicAdd(ap + 2, hv.z * coef);
    atomicAdd(ap + 3, hv.w * coef);
}

// ---------------- fused: x = gelu(LN(agg + h*dinv^2 + b)) ----------------
// wave32 per node, 2 features per lane, shuffle-butterfly reductions.
__global__ __launch_bounds__(256)
void gcn_ln_gelu_kernel(const float* __restrict__ agg, const float* __restrict__ h,
                        const float* __restrict__ dinv, const float* __restrict__ bias,
                        const float* __restrict__ g, const float* __restrict__ be,
                        float* __restrict__ xout) {
    int node = blockIdx.x * (blockDim.x >> 5) + (threadIdx.x >> 5);
    int lane = threadIdx.x & 31;
    if (node >= N_NODES) return;
    float di = dinv[node];
    float sc = di * di;
    size_t base = (size_t)node * HID;

    float v0 = agg[base + lane]      + h[base + lane]      * sc + bias[lane];
    float v1 = agg[base + lane + 32] + h[base + lane + 32] * sc + bias[lane + 32];

    float s = v0 + v1;
#pragma unroll
    for (int o = 16; o > 0; o >>= 1) s += __shfl_xor(s, o, 32);
    float mu = s * (1.0f / HID);

    float d0 = v0 - mu, d1 = v1 - mu;
    float vv = d0 * d0 + d1 * d1;
#pragma unroll
    for (int o = 16; o > 0; o >>= 1) vv += __shfl_xor(vv, o, 32);
    float rstd = rsqrtf(vv * (1.0f / HID) + 1e-5f);

    float y0 = d0 * rstd * g[lane]      + be[lane];
    float y1 = d1 * rstd * g[lane + 32] + be[lane + 32];
    // exact GELU: 0.5*x*(1+erf(x/sqrt(2)))
    y0 = 0.5f * y0 * (1.0f + erff(y0 * 0.70710678118654752440f));
    y1 = 0.5f * y1 * (1.0f + erff(y1 * 0.70710678118654752440f));

    xout[base + lane]      = y0;
    xout[base + lane + 32] = y1;
}

// ---------------- head: out[n] = dot(x[n,:], Wh) + bh ----------------
__global__ __launch_bounds__(256)
void gcn_head_kernel(const float* __restrict__ x, const float* __restrict__ Wh,
                     const float* __restrict__ bh, float* __restrict__ out) {
    int node = blockIdx.x * (blockDim.x >> 5) + (threadIdx.x >> 5);
    int lane = threadIdx.x & 31;
    if (node >= N_NODES) return;
    size_t base = (size_t)node * HID;
    float s = x[base + lane] * Wh[lane] + x[base + lane + 32] * Wh[lane + 32];
#pragma unroll
    for (int o = 16; o > 0; o >>= 1) s += __shfl_xor(s, o, 32);
    if (lane == 0) out[node] = s + bh[0];
}

extern "C" void kernel_launch(void* const* d_in, const int* in_sizes, int n_in,
                              void* d_out, int out_size, void* d_ws, size_t ws_size,
                              hipStream_t stream) {
    const float* x    = (const float*)d_in[0];
    const int*   ei   = (const int*)d_in[1];      // [2, E] int32
    const float* W1   = (const float*)d_in[2];
    const float* b1   = (const float*)d_in[3];
    const float* g1   = (const float*)d_in[4];
    const float* be1  = (const float*)d_in[5];
    const float* W2   = (const float*)d_in[6];
    const float* b2   = (const float*)d_in[7];
    const float* g2   = (const float*)d_in[8];
    const float* be2  = (const float*)d_in[9];
    const float* W3   = (const float*)d_in[10];
    const float* b3   = (const float*)d_in[11];
    const float* g3   = (const float*)d_in[12];
    const float* be3  = (const float*)d_in[13];
    const float* Wh   = (const float*)d_in[14];
    const float* bh   = (const float*)d_in[15];
    float* out = (float*)d_out;

    const int* src = ei;
    const int* dst = ei + N_EDGES;

    // workspace layout (floats): dinv | bufH | bufA | bufX  (~77.3 MB total)
    float* dinv = (float*)d_ws;                                    // N (deg -> dinv in place)
    float* bufH = dinv + (1 << 17);                                // N*HID
    float* bufA = bufH + (size_t)N_NODES * HID;                    // N*HID
    float* bufX = bufA + (size_t)N_NODES * HID;                    // N*HID

    const int TPB = 256;
    const int degBlocks   = (N_EDGES + TPB - 1) / TPB;             // 6250
    const int nodeBlocks  = (N_NODES + TPB - 1) / TPB;             // 391
    const int gemmBlocks  = (N_NODES + 127) / 128;                 // 782 (128 rows/block)
    const int scatBlocks  = (int)(((long long)N_EDGES * (HID / 4) + TPB - 1) / TPB); // 100000
    const int wnodeBlocks = (N_NODES + 7) / 8;                     // 12500 (8 waves/block)

    // degrees (self loop via +1 inside dinv kernel)
    hipMemsetAsync(dinv, 0, (size_t)N_NODES * sizeof(float), stream);
    gcn_deg_kernel<<<degBlocks, TPB, 0, stream>>>(dst, dinv);
    gcn_dinv_kernel<<<nodeBlocks, TPB, 0, stream>>>(dinv);

    // ---- layer 1 (K = 128) ----
    gcn_gemm_wmma_kernel<IN_DIM><<<gemmBlocks, TPB, 0, stream>>>(x, W1, bufH);
    hipMemsetAsync(bufA, 0, (size_t)N_NODES * HID * sizeof(float), stream);
    gcn_scatter_kernel<<<scatBlocks, TPB, 0, stream>>>(src, dst, dinv, bufH, bufA);
    gcn_ln_gelu_kernel<<<wnodeBlocks, TPB, 0, stream>>>(bufA, bufH, dinv, b1, g1, be1, bufX);

    // ---- layer 2 (K = 64) ----
    gcn_gemm_wmma_kernel<HID><<<gemmBlocks, TPB, 0, stream>>>(bufX, W2, bufH);
    hipMemsetAsync(bufA, 0, (size_t)N_NODES * HID * sizeof(float), stream);
    gcn_scatter_kernel<<<scatBlocks, TPB, 0, stream>>>(src, dst, dinv, bufH, bufA);
    gcn_ln_gelu_kernel<<<wnodeBlocks, TPB, 0, stream>>>(bufA, bufH, dinv, b2, g2, be2, bufX);

    // ---- layer 3 (K = 64) ----
    gcn_gemm_wmma_kernel<HID><<<gemmBlocks, TPB, 0, stream>>>(bufX, W3, bufH);
    hipMemsetAsync(bufA, 0, (size_t)N_NODES * HID * sizeof(float), stream);
    gcn_scatter_kernel<<<scatBlocks, TPB, 0, stream>>>(src, dst, dinv, bufH, bufA);
    gcn_ln_gelu_kernel<<<wnodeBlocks, TPB, 0, stream>>>(bufA, bufH, dinv, b3, g3, be3, bufX);

    // ---- head ----
    gcn_head_kernel<<<wnodeBlocks, TPB, 0, stream>>>(bufX, Wh, bh, out);

    (void)in_sizes; (void)n_in; (void)out_size; (void)ws_size;
}